// TorchVecEnv_20306605376168
// MI455X (gfx1250) — compile-verified
//
#include <hip/hip_runtime.h>
#include <stdint.h>

// CDNA5 / gfx1250 — one wave32-quad (128 threads) per environment.
// Stages a speculative 13x13 halo around the agent into LDS with
// GLOBAL_LOAD_ASYNC_TO_LDS_B32 (ASYNCcnt-tracked), then resolves the move and
// emits the 6x11x11 observation with coalesced channel stores.

namespace {
constexpr int H = 64, W = 64;
constexpr int VIEW = 11, RAD = 5;
constexpr int REG = 13;                   // halo tile: covers window for any |move|<=1
constexpr int REG_ELEMS = REG * REG;      // 169
constexpr int OBS_PIX = VIEW * VIEW;      // 121
constexpr int OBS_PER_ENV = 6 * OBS_PIX;  // 726
}

__global__ __launch_bounds__(128)
void forage_step_obs_kernel(const float* __restrict__ grids,
                            const float* __restrict__ energy_in,
                            const float* __restrict__ lut,
                            const int* __restrict__ agent_x,
                            const int* __restrict__ agent_y,
                            const int* __restrict__ actions,
                            const int* __restrict__ deltas,
                            float* __restrict__ out)
{
    __shared__ float tile[REG_ELEMS];
    __shared__ float slut[8];

    const int env = blockIdx.x;
    const int t   = (int)threadIdx.x;

    const int y   = agent_y[env];
    const int x   = agent_x[env];
    const int act = actions[env];

    if (t < 5) slut[t] = lut[t];

    const float* g = grids + (size_t)env * (H * W);
    const unsigned ldsBase = (unsigned)(uintptr_t)(&tile[0]); // low 32 bits = LDS byte offset

    // --- stage 13x13 clamped halo into LDS via async-tensor path ---
    #pragma unroll
    for (int it = 0; it < 2; ++it) {
        int idx = t + it * 128;
        idx = idx < REG_ELEMS ? idx : REG_ELEMS - 1;   // duplicate-safe clamp (same value)
        const int ry = idx / REG, rx = idx - ry * REG;
        int gy = y - 6 + ry; gy = gy < 0 ? 0 : (gy > H - 1 ? H - 1 : gy);
        int gx = x - 6 + rx; gx = gx < 0 ? 0 : (gx > W - 1 ? W - 1 : gx);
        const unsigned lds = ldsBase + (unsigned)idx * 4u;
        const uint64_t src = (uint64_t)(uintptr_t)(g + gy * W + gx);
        asm volatile("global_load_async_to_lds_b32 %0, %1, off"
                     :: "v"(lds), "v"(src)
                     : "memory");
    }
    asm volatile("s_wait_asynccnt 0" ::: "memory");   // this wave's async copies done
    __syncthreads();                                  // all waves' copies visible

    // --- resolve move (uniform per block; recomputed per lane from LDS) ---
    const int dy = deltas[act * 2 + 0];
    const int dx = deltas[act * 2 + 1];
    int nyc = y + dy; nyc = nyc < 0 ? 0 : (nyc > H - 1 ? H - 1 : nyc);
    int nxc = x + dx; nxc = nxc < 0 ? 0 : (nxc > W - 1 ? W - 1 : nxc);
    const float tgt = tile[(nyc - y + 6) * REG + (nxc - x + 6)];
    const bool blocked = (tgt == 1.0f);               // WALL blocks the move
    const int ny = blocked ? y : nyc;
    const int nx = blocked ? x : nxc;
    const float cell = tile[(ny - y + 6) * REG + (nx - x + 6)];
    const bool isF = (cell == 2.0f);
    const bool isP = (cell == 3.0f);
    const bool consumed = isF || isP;
    const float reward = (isF ? 10.0f : 0.0f) - (isP ? 10.0f : 0.0f);
    const float energy = energy_in[env] - 1.0f + reward;

    // --- emit one window pixel per lane, 6 coalesced channel stores ---
    if (t < OBS_PIX) {
        const int r = t / VIEW, c = t - r * VIEW;
        const int wy = ny - RAD + r;
        const int wx = nx - RAD + c;
        float v = 1.0f;                               // WALL padding outside grid
        if (wy >= 0 && wy < H && wx >= 0 && wx < W)
            v = tile[(wy - y + 6) * REG + (wx - x + 6)];
        if (r == RAD && c == RAD && consumed)
            v = 0.0f;                                 // consumed cell -> EMPTY
        float* o = out + (size_t)env * OBS_PER_ENV + t;
        o[0 * OBS_PIX] = (v == 2.0f) ? 1.0f : 0.0f;   // food mask
        o[1 * OBS_PIX] = (v == 3.0f) ? 1.0f : 0.0f;   // poison mask
        o[2 * OBS_PIX] = (v == 1.0f) ? 1.0f : 0.0f;   // wall mask
        o[3 * OBS_PIX] = slut[(int)v];                // interestingness LUT
        o[4 * OBS_PIX] = v;                           // raw window
        o[5 * OBS_PIX] = energy;                      // broadcast energy
    }
}

extern "C" void kernel_launch(void* const* d_in, const int* in_sizes, int n_in,
                              void* d_out, int out_size, void* d_ws, size_t ws_size,
                              hipStream_t stream)
{
    const float* grids  = (const float*)d_in[0];
    const float* energy = (const float*)d_in[1];
    const float* lut    = (const float*)d_in[2];
    const int*   ax     = (const int*)d_in[3];
    const int*   ay     = (const int*)d_in[4];
    // d_in[5] = agent_steps: unused by the reference output
    const int*   acts   = (const int*)d_in[6];
    const int*   dlt    = (const int*)d_in[7];
    const int n = in_sizes[1];  // N_ENVS
    forage_step_obs_kernel<<<n, 128, 0, stream>>>(grids, energy, lut, ax, ay,
                                                  acts, dlt, (float*)d_out);
}